// MockCFM_17179869193
// MI455X (gfx1250) — compile-verified
//
#include <hip/hip_runtime.h>

#define Bsz   8
#define Cch   128
#define Tlen  8192
#define SIGMA_MIN 1e-6f
#define BT    64        // timesteps per block chunk
#define PITCH 132       // LDS row pitch in floats (conflict-free B-frag reads)

typedef __attribute__((ext_vector_type(2))) float v2f;
typedef __attribute__((ext_vector_type(8))) float v8f;

__global__ __launch_bounds__(256) void cfm_main_kernel(
    const float* __restrict__ x1, const float* __restrict__ z,
    const float* __restrict__ W, const float* __restrict__ b_lin,
    const float* __restrict__ t_sample, const int* __restrict__ x_lens,
    const int* __restrict__ prompt_lens,
    float* __restrict__ out2,        // [B,C,T]
    float* __restrict__ partials)    // one float per block
{
  __shared__ __align__(16) float y_lds[BT * PITCH];  // y transposed: [t_local][c]
  __shared__ float red[256];

  const int b     = blockIdx.y;
  const int tBase = blockIdx.x * BT;
  const int tid   = threadIdx.x;
  const int lane  = tid & 31;
  const int wave  = tid >> 5;
  const int col   = lane & 15;        // N index (or A-frag M row)
  const int hi    = (lane >> 4) & 1;  // half-wave

  const float ts    = t_sample[b];
  const float omsig = 1.0f - SIGMA_MIN;
  const float acoef = 1.0f - omsig * ts;
  const int   pl    = prompt_lens[b];
  const int   xl    = x_lens[b];
  const float scale = 1.0f / ((float)Bsz * (float)Cch * (float)(xl - pl));

  // ---- stage y tile into LDS (transposed), b128 global loads ----
  // Per wave-instruction: 4 c-rows x 8 t-quads -> 4 full cachelines, coalesced.
  const size_t baseBT = (size_t)b * Cch * Tlen + (size_t)tBase;
  const int gc = lane >> 3;   // 0..3 : c offset within group
  const int q8 = lane & 7;    // 0..7 : t-quad within half row
  #pragma unroll
  for (int i = 0; i < 8; ++i) {
    int WT    = i * 8 + wave;        // 0..63 wave-task
    int c     = (WT >> 1) * 4 + gc;  // channel row
    int tl4   = ((WT & 1) * 8 + q8) * 4;  // t_local of quad start
    int t0    = tBase + tl4;
    size_t g  = baseBT + (size_t)c * Tlen + (size_t)tl4;
    float4 z4 = *(const float4*)&z[g];
    float4 x4 = *(const float4*)&x1[g];
    float y0 = (t0 + 0 >= pl) ? acoef * z4.x + ts * x4.x : 0.0f;
    float y1 = (t0 + 1 >= pl) ? acoef * z4.y + ts * x4.y : 0.0f;
    float y2 = (t0 + 2 >= pl) ? acoef * z4.z + ts * x4.z : 0.0f;
    float y3 = (t0 + 3 >= pl) ? acoef * z4.w + ts * x4.w : 0.0f;
    y_lds[(tl4 + 0) * PITCH + c] = y0;
    y_lds[(tl4 + 1) * PITCH + c] = y1;
    y_lds[(tl4 + 2) * PITCH + c] = y2;
    y_lds[(tl4 + 3) * PITCH + c] = y3;
  }
  __syncthreads();

  // ---- preload A fragments: rows d0..d0+15 of W, all K=128 ----
  // A 16x4 f32 layout: lane M = col; lanes<16 hold K={0,1}, lanes>=16 K={2,3}
  const int d0 = wave * 16;
  v2f a_frag[32];
  #pragma unroll
  for (int k = 0; k < 32; ++k) {
    a_frag[k] = *(const v2f*)&W[(d0 + col) * Cch + 4 * k + 2 * hi];
  }

  float loss_acc = 0.0f;

  // ---- each wave: one d-tile, 4 t-tiles per chunk ----
  for (int tt = 0; tt < BT / 16; ++tt) {
    const int tl0 = tt * 16;
    // bias folded into accumulator (C/D layout: VGPR v holds rows v and v+8)
    v8f acc;
    #pragma unroll
    for (int v = 0; v < 8; ++v) acc[v] = b_lin[d0 + v + 8 * hi];

    #pragma unroll
    for (int k = 0; k < 32; ++k) {
      // B 4x16 f32 layout: lane N = col; lanes<16 K={0,1}, lanes>=16 K={2,3}
      v2f bf = *(const v2f*)&y_lds[(tl0 + col) * PITCH + 4 * k + 2 * hi];
      acc = __builtin_amdgcn_wmma_f32_16x16x4_f32(
          /*neg_a=*/false, a_frag[k], /*neg_b=*/false, bf,
          /*c_mod=*/(short)0, acc, /*reuse_a=*/false, /*reuse_b=*/false);
    }

    const int  t  = tBase + tl0 + col;
    const bool lm = (t >= pl) && (t < xl);
    #pragma unroll
    for (int v = 0; v < 8; ++v) {
      int d = d0 + v + 8 * hi;
      size_t g = (size_t)(b * Cch + d) * Tlen + (size_t)t;
      float zv  = z[g];
      float x1v = x1[g];
      float o2  = acc[v] + omsig * zv;     // out + (1-sigma)*z
      out2[g] = o2;
      float diff = o2 - x1v;               // out - u == out2 - x1
      if (lm) loss_acc += diff * diff * scale;
    }
  }

  // ---- deterministic block reduction of loss ----
  red[tid] = loss_acc;
  __syncthreads();
  #pragma unroll
  for (int s = 128; s > 0; s >>= 1) {
    if (tid < s) red[tid] += red[tid + s];
    __syncthreads();
  }
  if (tid == 0) partials[blockIdx.y * gridDim.x + blockIdx.x] = red[0];
}

__global__ __launch_bounds__(256) void cfm_reduce_kernel(
    const float* __restrict__ partials, int n, float* __restrict__ loss_out)
{
  __shared__ float red[256];
  float s = 0.0f;
  for (int i = threadIdx.x; i < n; i += 256) s += partials[i];
  red[threadIdx.x] = s;
  __syncthreads();
  #pragma unroll
  for (int st = 128; st > 0; st >>= 1) {
    if (threadIdx.x < st) red[threadIdx.x] += red[threadIdx.x + st];
    __syncthreads();
  }
  if (threadIdx.x == 0) loss_out[0] = red[0];
}

extern "C" void kernel_launch(void* const* d_in, const int* in_sizes, int n_in,
                              void* d_out, int out_size, void* d_ws, size_t ws_size,
                              hipStream_t stream) {
  // setup_inputs order: x1, mu, style, t_sample, z, W, b_lin, x_lens, prompt_lens
  const float* x1          = (const float*)d_in[0];
  const float* t_sample    = (const float*)d_in[3];
  const float* z           = (const float*)d_in[4];
  const float* W           = (const float*)d_in[5];
  const float* b_lin       = (const float*)d_in[6];
  const int*   x_lens      = (const int*)d_in[7];
  const int*   prompt_lens = (const int*)d_in[8];

  float* out      = (float*)d_out;   // out[0] = loss; out[1..] = out2 [B,C,T]
  float* partials = (float*)d_ws;    // Tlen/BT * Bsz = 1024 floats

  dim3 grid(Tlen / BT, Bsz);
  cfm_main_kernel<<<grid, 256, 0, stream>>>(x1, z, W, b_lin, t_sample,
                                            x_lens, prompt_lens,
                                            out + 1, partials);
  cfm_reduce_kernel<<<1, 256, 0, stream>>>(partials, (Tlen / BT) * Bsz, out);
}